// MS_DGCNN2_Ablation_81870666596736
// MI455X (gfx1250) — compile-verified
//
#include <hip/hip_runtime.h>
#include <math.h>

typedef __attribute__((ext_vector_type(2))) float v2f;
typedef __attribute__((ext_vector_type(8))) float v8f;

#define NPTS 2048
#define KMAX 30

#define AS3 __attribute__((address_space(3)))

// Coalesced writer for one scale; K compile-time so /K, %K are mul-shift.
template<int K, int CH>
__device__ __forceinline__ void write_scale(float* __restrict__ ob, int b, int rwave,
                                            const int* __restrict__ idxL,
                                            const float* __restrict__ X,
                                            const float* __restrict__ Y,
                                            const float* __restrict__ Z,
                                            int lane) {
    for (int ch = 0; ch < CH; ++ch) {
        const long chbase = ((long)(b * CH + ch)) * NPTS;
#pragma unroll 1
        for (int t = lane; t < 32 * K; t += 32) {
            const int nl = t / K, j = t % K;
            const int n  = rwave + nl;
            const int id = idxL[nl * KMAX + j];
            const float cx = X[n], cy = Y[n], cz = Z[n];
            const float rx = X[id] - cx;
            const float ry = Y[id] - cy;
            const float rz = Z[id] - cz;
            float val;
            if (ch < 3) {
                val = (ch == 0) ? rx : ((ch == 1) ? ry : rz);
            } else if (CH == 9 && ch < 6) {
                const float ss  = rx * rx + ry * ry + rz * rz;
                const float nrm = (ss > 0.0f) ? sqrtf(ss) : 0.0f;
                const float inv = 1.0f / (nrm + 1e-8f);
                const int   c2  = ch - 3;
                val = ((c2 == 0) ? rx : ((c2 == 1) ? ry : rz)) * inv;
            } else {
                const int c2 = ch - (CH == 9 ? 6 : 3);
                val = (c2 == 0) ? cx : ((c2 == 1) ? cy : cz);
            }
            ob[(chbase + n) * K + j] = val;
        }
    }
}

__global__ __launch_bounds__(64) void msdgcnn_knn_feat(const float* __restrict__ xg,
                                                       float* __restrict__ out) {
    // SoA point stage: x[b] copied verbatim ([c][n] layout), |p|^2 separate.
    __shared__ __align__(16) float ptsXYZ[3 * NPTS];   // 24 KB
    __shared__ float sq[NPTS];                         // 8 KB
    __shared__ float innerS[2][32 * 17];               // per-wave Gram stage (pad 17)
    __shared__ int   idxS[2][32 * KMAX];               // per-wave top-30 indices

    const int tid   = threadIdx.x;
    const int lane  = tid & 31;
    const int wave  = tid >> 5;
    const int b     = blockIdx.x >> 5;                 // 32 blocks per batch
    const int nblk  = (blockIdx.x & 31) * 64;          // 64 rows per block
    const int rwave = nblk + wave * 32;                // 32 rows per wave

    const float* xb = xg + (size_t)b * 3 * NPTS;

    // ---- Stage the contiguous 24KB batch block into LDS via the gfx1250
    // async-to-LDS path (ASYNCcnt). Each lane moves one 16B chunk per iter.
    for (int i = tid; i < (3 * NPTS) / 4; i += 64) {
        const float* gp = xb + 4 * i;
        unsigned ldsAddr = (unsigned)(size_t)(AS3 float*)(ptsXYZ + 4 * i);
        asm volatile("global_load_async_to_lds_b128 %0, %1, off"
                     :
                     : "v"(ldsAddr), "v"(gp)
                     : "memory");
    }
    asm volatile("s_wait_asynccnt 0" ::: "memory");
    __syncthreads();

    // squared norms
    for (int n = tid; n < NPTS; n += 64) {
        const float px = ptsXYZ[n], py = ptsXYZ[NPTS + n], pz = ptsXYZ[2 * NPTS + n];
        sq[n] = px * px + py * py + pz * pz;
    }
    __syncthreads();

    const float* X = ptsXYZ;
    const float* Y = ptsXYZ + NPTS;
    const float* Z = ptsXYZ + 2 * NPTS;

    float* innerL = innerS[wave];
    int*   idxL   = idxS[wave];

    const int  m16 = lane & 15;
    const bool hi  = lane >= 16;

    // A operands (ISA 16x4 f32): lanes 0-15 K={0,1}=(x,y); lanes 16-31 K={2,3}=(z,0)
    const int r0 = rwave + m16;
    const int r1 = r0 + 16;
    v2f a0, a1;
    a0.x = hi ? Z[r0] : X[r0];
    a0.y = hi ? 0.0f  : Y[r0];
    a1.x = hi ? Z[r1] : X[r1];
    a1.y = hi ? 0.0f  : Y[r1];

    const float sqr = sq[rwave + lane];

    float bd[KMAX];
    int   bi[KMAX];
#pragma unroll
    for (int j = 0; j < KMAX; ++j) { bd[j] = -3.0e38f; bi[j] = 0; }

    const v8f czero = {};

    // Hoisted LDS base pointers -> ds ops with immediate offsets inside the loop
    float*       stBase = innerL + (hi ? 8 * 17 : 0) + m16;  // D scatter base
    const float* rdBase = innerL + lane * 17;                // selection read base
    const int    bxSel  = hi ? NPTS : 0;                     // bv.x: x-array or y-array

#pragma unroll 1
    for (int ct = 0; ct < NPTS / 16; ++ct) {
        const int colBase = ct * 16;
        const int col     = colBase + m16;
        // B (4x16 f32): VGPR0 lanes0-15 K=0 (x) / lanes16-31 K=1 (y); VGPR1 K=2 (z) / K=3 (0)
        v2f bv;
        bv.x = ptsXYZ[bxSel + col];
        const float zv = Z[col];
        bv.y = hi ? 0.0f : zv;
        const float sqc = sq[col];  // column |p|^2, folded into the staged value

        v8f d0 = __builtin_amdgcn_wmma_f32_16x16x4_f32(false, a0, false, bv,
                                                       (short)0, czero, false, false);
        v8f d1 = __builtin_amdgcn_wmma_f32_16x16x4_f32(false, a1, false, bv,
                                                       (short)0, czero, false, false);

        // Stage 2*inner - sq_col (lane = column, VGPR j = row j/j+8)
#pragma unroll
        for (int j = 0; j < 8; ++j) {
            stBase[j * 17]        = __builtin_fmaf(2.0f, d0[j], -sqc);
            stBase[(16 + j) * 17] = __builtin_fmaf(2.0f, d1[j], -sqc);
        }
        asm volatile("s_wait_dscnt 0" ::: "memory");

        // Each lane owns one row: pd = staged - sq_row; keep sorted top-30
#pragma unroll 1
        for (int c = 0; c < 16; ++c) {
            const float pd = rdBase[c] - sqr;
            const int   cc = colBase + c;
            if (pd > bd[KMAX - 1]) {
                bd[KMAX - 1] = pd;
                bi[KMAX - 1] = cc;
#pragma unroll
                for (int j = KMAX - 1; j > 0; --j) {
                    if (bd[j] > bd[j - 1]) {
                        float td = bd[j]; bd[j] = bd[j - 1]; bd[j - 1] = td;
                        int   ti = bi[j]; bi[j] = bi[j - 1]; bi[j - 1] = ti;
                    }
                }
            }
        }
        // No wait needed here: same-wave DS ops are processed in order (reads of
        // this tile complete before next tile's stores).
    }

    // ---- Stage indices so the output phase is lane-coalesced ----
#pragma unroll
    for (int j = 0; j < KMAX; ++j) idxL[lane * KMAX + j] = bi[j];
    asm volatile("s_wait_dscnt 0" ::: "memory");

    // ---- Output: [B,6,N,5], [B,9,N,20], [B,9,N,30] ----
    write_scale<5, 6>(out, b, rwave, idxL, X, Y, Z, lane);
    const long base1 = (long)16 * 6 * NPTS * 5;
    const long base2 = base1 + (long)16 * 9 * NPTS * 20;
    write_scale<20, 9>(out + base1, b, rwave, idxL, X, Y, Z, lane);
    write_scale<30, 9>(out + base2, b, rwave, idxL, X, Y, Z, lane);
}

extern "C" void kernel_launch(void* const* d_in, const int* in_sizes, int n_in,
                              void* d_out, int out_size, void* d_ws, size_t ws_size,
                              hipStream_t stream) {
    (void)in_sizes; (void)n_in; (void)out_size; (void)d_ws; (void)ws_size;
    const float* x = (const float*)d_in[0];
    float* out     = (float*)d_out;
    dim3 grid(512);   // 16 batches * 32 row-blocks (64 rows each)
    dim3 block(64);   // 2 waves of 32; each wave owns 32 rows
    hipLaunchKernelGGL(msdgcnn_knn_feat, grid, block, 0, stream, x, out);
}